// GATLayer_1700807049605
// MI455X (gfx1250) — compile-verified
//
#include <hip/hip_runtime.h>
#include <math.h>

// GAT layer: B=32768, N=12, F=256, D=128
// out = elu( softmax_j( mask(leaky_relu(s_i + s_j)) ) @ H ),  H = X @ W
//
// Inputs (setup_inputs order): x[B,N,F] f32, adj[N,N] f32, W[F,D] f32, a[2D,1] f32
// Output: [B,N,D] f32

#define GAT_B 32768
#define GAT_N 12
#define GAT_F 256
#define GAT_D 128

#define BATCHES_PER_BLOCK 16                 // 16 batches * 12 nodes = 192 rows
#define ROWS_PER_BLOCK (BATCHES_PER_BLOCK * GAT_N)  // 192
#define WAVES_PER_BLOCK (ROWS_PER_BLOCK / 16)       // 12 waves, 16 rows each
#define THREADS_PER_BLOCK (WAVES_PER_BLOCK * 32)    // 384
#define HSTR 132                              // padded LDS row stride (floats), 132%64 != 0 banks

typedef __attribute__((ext_vector_type(2))) float v2f;
typedef __attribute__((ext_vector_type(8))) float v8f;

__global__ __launch_bounds__(THREADS_PER_BLOCK)
void gat_fused_kernel(const float* __restrict__ x,
                      const float* __restrict__ adj,
                      const float* __restrict__ Wm,
                      const float* __restrict__ avec,
                      float* __restrict__ out) {
    __shared__ float Hlds[ROWS_PER_BLOCK * HSTR];          // 192*132*4 = 101376 B
    __shared__ float alpha_lds[ROWS_PER_BLOCK * GAT_N];    // 192*12*4  = 9216 B
    __shared__ float s_i_lds[ROWS_PER_BLOCK];
    __shared__ float s_j_lds[ROWS_PER_BLOCK];

    const int tid   = threadIdx.x;
    const int wave  = tid >> 5;
    const int lane  = tid & 31;
    const int lhalf = lane >> 4;   // 0: lanes 0-15, 1: lanes 16-31
    const int lmod  = lane & 15;

    // ---------------- Phase 1: H = X @ W via V_WMMA_F32_16X16X4_F32 -------------
    // Wave `wave` computes global rows [g0, g0+16) of the flattened (B*N)xF GEMM.
    const long g0   = (long)blockIdx.x * ROWS_PER_BLOCK + wave * 16;
    const float* xrow = x + (g0 + lmod) * GAT_F;   // this lane's A-matrix row

    v8f acc[8];
    #pragma unroll
    for (int nt = 0; nt < 8; ++nt) {
        v8f z = {0.f, 0.f, 0.f, 0.f, 0.f, 0.f, 0.f, 0.f};
        acc[nt] = z;
    }

    #pragma unroll 4
    for (int ks = 0; ks < GAT_F / 4; ++ks) {
        // A-frag (16x4 f32): lanes 0-15 hold K={4ks,4ks+1}, lanes 16-31 hold K={4ks+2,4ks+3}
        const int k = ks * 4 + 2 * lhalf;
        const float2 av = *reinterpret_cast<const float2*>(xrow + k);
        v2f afrag;
        afrag.x = av.x;
        afrag.y = av.y;

        const float* wk = Wm + (long)k * GAT_D + lmod;
        #pragma unroll
        for (int nt = 0; nt < 8; ++nt) {
            // B-frag (4x16 f32): vgpr0 = W[k][col], vgpr1 = W[k+1][col] (k already lane-offset)
            v2f bfrag;
            bfrag.x = wk[nt * 16];
            bfrag.y = wk[nt * 16 + GAT_D];
            acc[nt] = __builtin_amdgcn_wmma_f32_16x16x4_f32(
                /*neg_a=*/false, afrag, /*neg_b=*/false, bfrag,
                /*c_mod=*/(short)0, acc[nt], /*reuse_a=*/false, /*reuse_b=*/false);
        }
    }

    // Spill H tile to LDS.  C/D layout: vgpr v holds M=v (lanes 0-15), M=v+8 (lanes 16-31)
    #pragma unroll
    for (int nt = 0; nt < 8; ++nt) {
        #pragma unroll
        for (int v = 0; v < 8; ++v) {
            const int m = v + 8 * lhalf;
            Hlds[(wave * 16 + m) * HSTR + nt * 16 + lmod] = acc[nt][v];
        }
    }
    __syncthreads();

    // ---------------- Phase 2: s_j = H . a[:D], s_i = H . a[D:] ------------------
    if (tid < ROWS_PER_BLOCK) {
        const float* hr = &Hlds[tid * HSTR];
        float sj = 0.f, si = 0.f;
        #pragma unroll 4
        for (int c = 0; c < GAT_D; c += 4) {
            const float4 h  = *reinterpret_cast<const float4*>(hr + c);
            const float4 a0 = *reinterpret_cast<const float4*>(avec + c);
            const float4 a1 = *reinterpret_cast<const float4*>(avec + GAT_D + c);
            sj += h.x * a0.x + h.y * a0.y + h.z * a0.z + h.w * a0.w;
            si += h.x * a1.x + h.y * a1.y + h.z * a1.z + h.w * a1.w;
        }
        s_j_lds[tid] = sj;
        s_i_lds[tid] = si;
    }
    __syncthreads();

    // ---------------- Phase 3a: masked leaky-relu + softmax over neighbors ------
    if (tid < ROWS_PER_BLOCK) {
        const int bb = tid / GAT_N;       // local batch
        const int i  = tid % GAT_N;       // node i
        const float si = s_i_lds[bb * GAT_N + i];
        float ev[GAT_N];
        float emax = -1e30f;
        #pragma unroll
        for (int j = 0; j < GAT_N; ++j) {
            float e = si + s_j_lds[bb * GAT_N + j];
            e = (e > 0.f) ? e : 0.2f * e;                       // leaky_relu(0.2)
            e = (adj[i * GAT_N + j] == 1.0f) ? e : -1e20f;      // mask
            ev[j] = e;
            emax = fmaxf(emax, e);
        }
        float sum = 0.f;
        #pragma unroll
        for (int j = 0; j < GAT_N; ++j) {
            ev[j] = __expf(ev[j] - emax);
            sum += ev[j];
        }
        const float inv = 1.0f / sum;
        #pragma unroll
        for (int j = 0; j < GAT_N; ++j)
            alpha_lds[tid * GAT_N + j] = ev[j] * inv;
    }
    __syncthreads();

    // ---------------- Phase 3b: out = elu(alpha @ H) -----------------------------
    // thread -> (row r = tid/2, d-half = (tid&1)*64); 64 outputs each, float4 I/O.
    {
        const int r  = tid >> 1;                  // 0..191
        const int d0 = (tid & 1) * (GAT_D / 2);   // 0 or 64
        const int bb = r / GAT_N;
        float al[GAT_N];
        #pragma unroll
        for (int j = 0; j < GAT_N; ++j) al[j] = alpha_lds[r * GAT_N + j];

        const int hbase = bb * GAT_N * HSTR;
        float* orow = out + ((long)blockIdx.x * ROWS_PER_BLOCK + r) * GAT_D;

        for (int d = d0; d < d0 + GAT_D / 2; d += 4) {
            float sx = 0.f, sy = 0.f, sz = 0.f, sw = 0.f;
            #pragma unroll
            for (int j = 0; j < GAT_N; ++j) {
                const float4 h = *reinterpret_cast<const float4*>(&Hlds[hbase + j * HSTR + d]);
                sx += al[j] * h.x;
                sy += al[j] * h.y;
                sz += al[j] * h.z;
                sw += al[j] * h.w;
            }
            float4 o;
            o.x = (sx > 0.f) ? sx : __expf(sx) - 1.0f;   // elu
            o.y = (sy > 0.f) ? sy : __expf(sy) - 1.0f;
            o.z = (sz > 0.f) ? sz : __expf(sz) - 1.0f;
            o.w = (sw > 0.f) ? sw : __expf(sw) - 1.0f;
            *reinterpret_cast<float4*>(orow + d) = o;
        }
    }
}

extern "C" void kernel_launch(void* const* d_in, const int* in_sizes, int n_in,
                              void* d_out, int out_size, void* d_ws, size_t ws_size,
                              hipStream_t stream) {
    const float* x    = (const float*)d_in[0];   // [B, N, F]
    const float* adj  = (const float*)d_in[1];   // [N, N]
    const float* Wm   = (const float*)d_in[2];   // [F, D]
    const float* avec = (const float*)d_in[3];   // [2D, 1]
    float* out        = (float*)d_out;           // [B, N, D]

    const int nblocks = GAT_B / BATCHES_PER_BLOCK;   // 2048
    gat_fused_kernel<<<dim3(nblocks), dim3(THREADS_PER_BLOCK), 0, stream>>>(
        x, adj, Wm, avec, out);
}